// Human3DKinematicLayer_40278203301929
// MI455X (gfx1250) — compile-verified
//
#include <hip/hip_runtime.h>
#include <cstdint>

#define WAVES_PER_BLOCK 4
#define BLOCK (WAVES_PER_BLOCK * 32)
#define E_STRIDE 52   // dwords/sample in LDS euler region (48 data + 4 pad; gcd(52,64)=4)
#define B_STRIDE 20   // dwords/sample in LDS bone region  (16 data + 4 pad)
#define O_STRIDE 51   // dwords/sample in LDS out region (unpadded: 51 odd -> conflict-free,
                      // and region is the exact contiguous global image for the wave)

// Low 32 bits of a generic pointer to LDS == group-segment byte offset (what
// DS / async-to/from-LDS instructions consume as the LDS address VGPR).
__device__ __forceinline__ unsigned lds_off(const void* p) {
  return (unsigned)(uintptr_t)p;
}

// CDNA5 async copies, tracked by ASYNCcnt.
__device__ __forceinline__ void async_ld_b128(unsigned lds_byte, const float* gaddr) {
  asm volatile("global_load_async_to_lds_b128 %0, %1, off"
               :: "v"(lds_byte), "v"((unsigned long long)(uintptr_t)gaddr)
               : "memory");
}
__device__ __forceinline__ void async_st_b128(float* gaddr, unsigned lds_byte) {
  asm volatile("global_store_async_from_lds_b128 %0, %1, off"
               :: "v"((unsigned long long)(uintptr_t)gaddr), "v"(lds_byte)
               : "memory");
}
__device__ __forceinline__ void async_st_b64(float* gaddr, unsigned lds_byte) {
  asm volatile("global_store_async_from_lds_b64 %0, %1, off"
               :: "v"((unsigned long long)(uintptr_t)gaddr), "v"(lds_byte)
               : "memory");
}
__device__ __forceinline__ void async_st_b32(float* gaddr, unsigned lds_byte) {
  asm volatile("global_store_async_from_lds_b32 %0, %1, off"
               :: "v"((unsigned long long)(uintptr_t)gaddr), "v"(lds_byte)
               : "memory");
}
__device__ __forceinline__ void wait_async0() {
  asm volatile("s_wait_asynccnt 0" ::: "memory");
}
__device__ __forceinline__ void wait_ds0() {
  asm volatile("s_wait_dscnt 0" ::: "memory");
}

// One kinematic bone: R = Rx(ex)@Ry(ey)@Rz(ez); M = M @ R; p += L * M.row2; emit p.
__device__ __forceinline__ void bone_step(float ex, float ey, float ez, float L,
                                          float M[9], float p[3],
                                          float* __restrict__ outJ) {
  float sx, cx, sy, cy, sz, cz;
  __sincosf(ex, &sx, &cx);
  __sincosf(ey, &sy, &cy);
  __sincosf(ez, &sz, &cz);
  const float sxsy = sx * sy;
  const float cxsy = cx * sy;
  const float R0 = cy * cz;
  const float R1 = -cy * sz;
  const float R2 = sy;
  const float R3 = fmaf(sxsy, cz, cx * sz);
  const float R4 = fmaf(-sxsy, sz, cx * cz);
  const float R5 = -sx * cy;
  const float R6 = fmaf(-cxsy, cz, sx * sz);
  const float R7 = fmaf(cxsy, sz, sx * cz);
  const float R8 = cx * cy;

  const float t0 = fmaf(M[0], R0, fmaf(M[1], R3, M[2] * R6));
  const float t1 = fmaf(M[0], R1, fmaf(M[1], R4, M[2] * R7));
  const float t2 = fmaf(M[0], R2, fmaf(M[1], R5, M[2] * R8));
  const float t3 = fmaf(M[3], R0, fmaf(M[4], R3, M[5] * R6));
  const float t4 = fmaf(M[3], R1, fmaf(M[4], R4, M[5] * R7));
  const float t5 = fmaf(M[3], R2, fmaf(M[4], R5, M[5] * R8));
  const float t6 = fmaf(M[6], R0, fmaf(M[7], R3, M[8] * R6));
  const float t7 = fmaf(M[6], R1, fmaf(M[7], R4, M[8] * R7));
  const float t8 = fmaf(M[6], R2, fmaf(M[7], R5, M[8] * R8));
  M[0] = t0; M[1] = t1; M[2] = t2;
  M[3] = t3; M[4] = t4; M[5] = t5;
  M[6] = t6; M[7] = t7; M[8] = t8;

  p[0] = fmaf(L, t6, p[0]);
  p[1] = fmaf(L, t7, p[1]);
  p[2] = fmaf(L, t8, p[2]);
  outJ[0] = p[0]; outJ[1] = p[1]; outJ[2] = p[2];
}

__device__ __forceinline__ void set_ident(float M[9], float p[3]) {
  M[0] = 1.f; M[1] = 0.f; M[2] = 0.f;
  M[3] = 0.f; M[4] = 1.f; M[5] = 0.f;
  M[6] = 0.f; M[7] = 0.f; M[8] = 1.f;
  p[0] = 0.f; p[1] = 0.f; p[2] = 0.f;
}

__global__ void __launch_bounds__(BLOCK)
fk_h36m_kernel(const float* __restrict__ euler,
               const float* __restrict__ bone,
               float* __restrict__ out, int N) {
  __shared__ __align__(16) float sE[WAVES_PER_BLOCK][32 * E_STRIDE];
  __shared__ __align__(16) float sB[WAVES_PER_BLOCK][32 * B_STRIDE];
  __shared__ __align__(16) float sO[WAVES_PER_BLOCK][32 * O_STRIDE];

  const int wave = threadIdx.x >> 5;
  const int lane = threadIdx.x & 31;
  const long long waveBase = (long long)blockIdx.x * BLOCK + wave * 32;
  if (waveBase >= N) return;  // N % 32 == 0 -> wave-uniform, no divergence

  // ---- Stage inputs: coalesced 512B async b128 passes, scattered into padded LDS.
  {
    const unsigned eb = lds_off(&sE[wave][0]);
    const float* g = euler + (size_t)waveBase * 48;
#pragma unroll
    for (int ps = 0; ps < 12; ++ps) {
      unsigned flat = (unsigned)(ps * 128) + (unsigned)lane * 4u;
      unsigned s = flat / 48u;               // sample within wave
      unsigned k = flat - s * 48u;           // float within sample (multiple of 4)
      async_ld_b128(eb + (s * E_STRIDE + k) * 4u, g + flat);
    }
    const unsigned bb = lds_off(&sB[wave][0]);
    const float* gb = bone + (size_t)waveBase * 16;
#pragma unroll
    for (int ps = 0; ps < 4; ++ps) {
      unsigned flat = (unsigned)(ps * 128) + (unsigned)lane * 4u;
      unsigned s = flat >> 4;
      unsigned k = flat & 15u;
      async_ld_b128(bb + (s * B_STRIDE + k) * 4u, gb + flat);
    }
  }
  wait_async0();  // all LDS writes for this wave landed

  const float* E  = &sE[wave][lane * E_STRIDE];  // 48 euler floats, bone i at E[3i..]
  const float* Lb = &sB[wave][lane * B_STRIDE];  // 16 bone lengths
  float* o = &sO[wave][lane * O_STRIDE];         // 51 output floats (17 joints x 3)

  // Root joint
  o[0] = 0.f; o[1] = 0.f; o[2] = 0.f;

  float M[9], p[3], M8[9], p8[3];

  // Chain A: root -> joints 1,2,3 (bones 0..2)
  set_ident(M, p);
  bone_step(E[0],  E[1],  E[2],  Lb[0], M, p, o + 3);
  bone_step(E[3],  E[4],  E[5],  Lb[1], M, p, o + 6);
  bone_step(E[6],  E[7],  E[8],  Lb[2], M, p, o + 9);

  // Chain B: root -> joints 4,5,6 (bones 3..5)
  set_ident(M, p);
  bone_step(E[9],  E[10], E[11], Lb[3], M, p, o + 12);
  bone_step(E[12], E[13], E[14], Lb[4], M, p, o + 15);
  bone_step(E[15], E[16], E[17], Lb[5], M, p, o + 18);

  // Chain C: root -> joints 7,8,9,10 (bones 6..9); save state at joint 8
  set_ident(M, p);
  bone_step(E[18], E[19], E[20], Lb[6], M, p, o + 21);  // joint 7
  bone_step(E[21], E[22], E[23], Lb[7], M, p, o + 24);  // joint 8
#pragma unroll
  for (int i = 0; i < 9; ++i) M8[i] = M[i];
  p8[0] = p[0]; p8[1] = p[1]; p8[2] = p[2];
  bone_step(E[24], E[25], E[26], Lb[8], M, p, o + 27);  // joint 9
  bone_step(E[27], E[28], E[29], Lb[9], M, p, o + 30);  // joint 10

  // Chain D: joint 8 -> joints 11,12,13 (bones 10..12)
#pragma unroll
  for (int i = 0; i < 9; ++i) M[i] = M8[i];
  p[0] = p8[0]; p[1] = p8[1]; p[2] = p8[2];
  bone_step(E[30], E[31], E[32], Lb[10], M, p, o + 33);
  bone_step(E[33], E[34], E[35], Lb[11], M, p, o + 36);
  bone_step(E[36], E[37], E[38], Lb[12], M, p, o + 39);

  // Chain E: joint 8 -> joints 14,15,16 (bones 13..15)
#pragma unroll
  for (int i = 0; i < 9; ++i) M[i] = M8[i];
  p[0] = p8[0]; p[1] = p8[1]; p[2] = p8[2];
  bone_step(E[39], E[40], E[41], Lb[13], M, p, o + 42);
  bone_step(E[42], E[43], E[44], Lb[14], M, p, o + 45);
  bone_step(E[45], E[46], E[47], Lb[15], M, p, o + 48);

  // ---- Drain: the wave's sO region is the exact contiguous 6528B global image.
  wait_ds0();  // ASYNC ops are unordered vs DS; ensure LDS writes are visible
  {
    const unsigned ob = lds_off(&sO[wave][0]);
    float* g = out + (size_t)waveBase * 51;   // byte offset 6528*m -> 16B aligned
#pragma unroll
    for (int ps = 0; ps < 12; ++ps) {         // dwords 0..1535, 512B per pass
      unsigned flat = (unsigned)(ps * 128) + (unsigned)lane * 4u;
      async_st_b128(g + flat, ob + flat * 4u);
    }
    {
      unsigned flat = 1536u + (unsigned)lane * 2u;  // dwords 1536..1599
      async_st_b64(g + flat, ob + flat * 4u);
    }
    {
      unsigned flat = 1600u + (unsigned)lane;       // dwords 1600..1631
      async_st_b32(g + flat, ob + flat * 4u);
    }
  }
  wait_async0();  // insurance; s_endpgm also does an implicit wait-idle
}

extern "C" void kernel_launch(void* const* d_in, const int* in_sizes, int n_in,
                              void* d_out, int out_size, void* d_ws, size_t ws_size,
                              hipStream_t stream) {
  const float* euler = (const float*)d_in[0];  // (N,16,3) f32
  const float* bone  = (const float*)d_in[1];  // (N,16,1) f32
  float* out = (float*)d_out;                  // (N,17,3) f32
  const int N = in_sizes[0] / 48;
  const int grid = (N + BLOCK - 1) / BLOCK;
  hipLaunchKernelGGL(fk_h36m_kernel, dim3(grid), dim3(BLOCK), 0, stream,
                     euler, bone, out, N);
}